// tPCN_26001732010585
// MI455X (gfx1250) — compile-verified
//
#include <hip/hip_runtime.h>
#include <math.h>

#define B_TOTAL   131072
#define H_DIM     64
#define NOBS      32
#define NDIR      8
#define MAX_ITER  10
#define LR_ITER   1e-3f
#define TOL       1e-3f
#define EPS_N     1e-6f

#define NTHREADS        256
#define ROWS_PER_BLOCK  128
#define NBLOCKS         (B_TOTAL / ROWS_PER_BLOCK)   // 1024

// ---- LDS layout (floats), strides padded for bank-conflict-free A-operand loads
#define WR_STR   68
#define WO_STR   68
#define WOT_STR  36
#define WIN_STR  9
#define OFF_WR   0
#define OFF_WO   (OFF_WR  + 64*WR_STR)    // 4352
#define OFF_WOT  (OFF_WO  + 32*WO_STR)    // 6528
#define OFF_WIN  (OFF_WOT + 64*WOT_STR)   // 8832
#define OFF_RED  (OFF_WIN + 64*WIN_STR)   // 9408
#define LDS_TOT  (OFF_RED + 8)            // 9416 floats = 37.7 KB

typedef __attribute__((ext_vector_type(2))) float v2f;
typedef __attribute__((ext_vector_type(8))) float v8f;

// ---------------------------------------------------------------------------
// State tiles live per-wave in WMMA C/D layout, transposed (N = 16 batch rows,
// M = feature dim). Lane L: b = L&15 (batch row), h = L>>4.
// xd[t][v] = X[row b][feature 16*t + 8*h + v].
// ---------------------------------------------------------------------------

__device__ __forceinline__ void load_state(const float* g, int r, int h, float xd[4][8]) {
#pragma unroll
  for (int t = 0; t < 4; ++t) {
    const float4* p4 = (const float4*)(g + (size_t)r * H_DIM + 16*t + 8*h);
    float4 a = p4[0], b = p4[1];
    xd[t][0]=a.x; xd[t][1]=a.y; xd[t][2]=a.z; xd[t][3]=a.w;
    xd[t][4]=b.x; xd[t][5]=b.y; xd[t][6]=b.z; xd[t][7]=b.w;
  }
}

__device__ __forceinline__ void store_state(float* g, int r, int h, const float xd[4][8]) {
#pragma unroll
  for (int t = 0; t < 4; ++t) {
    float4* p4 = (float4*)(g + (size_t)r * H_DIM + 16*t + 8*h);
    p4[0] = make_float4(xd[t][0], xd[t][1], xd[t][2], xd[t][3]);
    p4[1] = make_float4(xd[t][4], xd[t][5], xd[t][6], xd[t][7]);
  }
}

// Convert D-layout state -> WMMA B-operand chunks (K dim = NT*16 features).
// B operand chunk k, vgpr j: lane (b,h) must hold X[b][4k + 2h + j].
// Source value lives in tile t=k>>2, reg v=4(k&1)+2h+j, in lane-half h'=(k>>1)&1.
// One shuffle + one select per element.
template<int NT>
__device__ __forceinline__ void d2b(const float xd[NT][8], v2f* bb, int h) {
#pragma unroll
  for (int k = 0; k < NT*4; ++k) {
    const int t    = k >> 2;
    const int hp   = (k >> 1) & 1;
    const int base = 4 * (k & 1);
    float o0 = xd[t][base + 2*hp + 0];
    float o1 = xd[t][base + 2*hp + 1];
    float s0 = __shfl_xor(xd[t][base + 2*(1-hp) + 0], 16, 32);
    float s1 = __shfl_xor(xd[t][base + 2*(1-hp) + 1], 16, 32);
    v2f rr;
    rr[0] = (h == hp) ? o0 : s0;
    rr[1] = (h == hp) ? o1 : s1;
    bb[k] = rr;
  }
}

// acc[m] (MT tiles of 16 output features) += W(A, from LDS) x B(bb, KC chunks)
template<int MT, int KC>
__device__ __forceinline__ void wmma_mm(const float* Wl, int stride, const v2f* bb,
                                        v8f* acc, int p, int h) {
#pragma unroll
  for (int k = 0; k < KC; ++k) {
#pragma unroll
    for (int m = 0; m < MT; ++m) {
      v2f a = *(const v2f*)(Wl + (16*m + p) * stride + 4*k + 2*h);
      acc[m] = __builtin_amdgcn_wmma_f32_16x16x4_f32(false, a, false, bb[k],
                                                     (short)0, acc[m], false, false);
    }
  }
}

__device__ __forceinline__ void block_accum(float dsum, float* lds, float* partials) {
#pragma unroll
  for (int off = 16; off > 0; off >>= 1) dsum += __shfl_xor(dsum, off, 32);
  int wave = threadIdx.x >> 5;
  int lane = threadIdx.x & 31;
  if (lane == 0) lds[OFF_RED + wave] = dsum;
  __syncthreads();
  if (threadIdx.x == 0) {
    float t = 0.f;
#pragma unroll
    for (int i = 0; i < 8; ++i) t += lds[OFF_RED + i];
    partials[blockIdx.x] = t;
  }
}

// ---------------------------------------------------------------------------
__global__ void __launch_bounds__(NTHREADS) tpcn_init(
    const int* __restrict__ dirs, const float* __restrict__ prev_states,
    const float* __restrict__ Wr, const float* __restrict__ Win,
    float* __restrict__ sbuf, float* __restrict__ pbuf, float* __restrict__ partials)
{
  __shared__ float lds[LDS_TOT];
  for (int i = threadIdx.x; i < 64*64; i += NTHREADS) lds[OFF_WR  + (i>>6)*WR_STR  + (i&63)] = Wr[i];
  for (int i = threadIdx.x; i < 64*8;  i += NTHREADS) lds[OFF_WIN + (i>>3)*WIN_STR + (i&7)]  = Win[i];
  __syncthreads();

  int wave = threadIdx.x >> 5, lane = threadIdx.x & 31;
  int b = lane & 15, h = lane >> 4;
  int r = blockIdx.x * ROWS_PER_BLOCK + wave * 16 + b;

  float pd[4][8];
  load_state(prev_states, r, h, pd);
  float ssq = 0.f;
#pragma unroll
  for (int t = 0; t < 4; ++t)
#pragma unroll
    for (int v = 0; v < 8; ++v) ssq += pd[t][v] * pd[t][v];
  ssq += __shfl_xor(ssq, 16, 32);
  float sc = 1.f / (sqrtf(ssq) + EPS_N);
#pragma unroll
  for (int t = 0; t < 4; ++t)
#pragma unroll
    for (int v = 0; v < 8; ++v) pd[t][v] *= sc;

  int dir = dirs[r];
  v8f acc[4];
#pragma unroll
  for (int t = 0; t < 4; ++t)
#pragma unroll
    for (int v = 0; v < 8; ++v)
      acc[t][v] = lds[OFF_WIN + (16*t + 8*h + v) * WIN_STR + dir];   // drive

  v2f bb[16];
  d2b<4>(pd, bb, h);
  wmma_mm<4, 16>(lds + OFF_WR, WR_STR, bb, acc, b, h);               // + Wr @ prev^T

  float sd[4][8];
  float ssq2 = 0.f;
#pragma unroll
  for (int t = 0; t < 4; ++t)
#pragma unroll
    for (int v = 0; v < 8; ++v) { float g = fmaxf(acc[t][v], 0.f); sd[t][v] = g; ssq2 += g*g; }
  ssq2 += __shfl_xor(ssq2, 16, 32);
  float sc2 = 1.f / (sqrtf(ssq2) + EPS_N);
  float dsum = 0.f;
#pragma unroll
  for (int t = 0; t < 4; ++t)
#pragma unroll
    for (int v = 0; v < 8; ++v) { float x = sd[t][v] * sc2; sd[t][v] = x; dsum += pd[t][v] - x; }

  store_state(pbuf, r, h, pd);
  store_state(sbuf, r, h, sd);
  block_accum(dsum, lds, partials);
}

// ---------------------------------------------------------------------------
__global__ void __launch_bounds__(NTHREADS) tpcn_flag(
    float* __restrict__ meta, const float* __restrict__ partials, int it, int first)
{
  __shared__ float red[NTHREADS];
  float a = 0.f;
  for (int i = threadIdx.x; i < NBLOCKS; i += NTHREADS) a += partials[i];
  red[threadIdx.x] = a;
  __syncthreads();
  for (int s = NTHREADS / 2; s > 0; s >>= 1) {
    if (threadIdx.x < s) red[threadIdx.x] += red[threadIdx.x + s];
    __syncthreads();
  }
  if (threadIdx.x == 0) {
    float done = first ? 0.f : meta[0];
    float mean = red[0] * (1.0f / (float)(B_TOTAL * H_DIM));
    bool active = (done == 0.f) && (fabsf(mean) > TOL);
    meta[0]      = active ? 0.f : 1.f;   // done = !active (latches)
    meta[1 + it] = active ? 1.f : 0.f;
  }
}

// ---------------------------------------------------------------------------
__global__ void __launch_bounds__(NTHREADS) tpcn_step(
    const float* __restrict__ meta, int it,
    float* __restrict__ sbuf, float* __restrict__ pbuf,
    const int* __restrict__ dirs, const int* __restrict__ obs,
    const float* __restrict__ Wr, const float* __restrict__ Wout, const float* __restrict__ Win,
    float* __restrict__ partials)
{
  if (meta[1 + it] == 0.0f) return;   // grid-uniform: inactive => state passes through
  __shared__ float lds[LDS_TOT];
  for (int i = threadIdx.x; i < 64*64; i += NTHREADS) lds[OFF_WR  + (i>>6)*WR_STR  + (i&63)] = Wr[i];
  for (int i = threadIdx.x; i < 32*64; i += NTHREADS) lds[OFF_WO  + (i>>6)*WO_STR  + (i&63)] = Wout[i];
  for (int i = threadIdx.x; i < 64*32; i += NTHREADS) { int hh = i>>5, o = i&31;
    lds[OFF_WOT + hh*WOT_STR + o] = Wout[o*64 + hh]; }
  for (int i = threadIdx.x; i < 64*8;  i += NTHREADS) lds[OFF_WIN + (i>>3)*WIN_STR + (i&7)]  = Win[i];
  __syncthreads();

  int wave = threadIdx.x >> 5, lane = threadIdx.x & 31;
  int b = lane & 15, h = lane >> 4;
  int r = blockIdx.x * ROWS_PER_BLOCK + wave * 16 + b;

  float sd[4][8], pd[4][8];
  load_state(sbuf, r, h, sd);
  load_state(pbuf, r, h, pd);
  int dir = dirs[r];
  int ob  = obs[r];

  // acc = drive; then acc = relu(Wr@prev^T + drive) - s   (= -eps_g)
  v8f acc[4];
#pragma unroll
  for (int t = 0; t < 4; ++t)
#pragma unroll
    for (int v = 0; v < 8; ++v)
      acc[t][v] = lds[OFF_WIN + (16*t + 8*h + v) * WIN_STR + dir];

  v2f bb[16];
  d2b<4>(pd, bb, h);
  wmma_mm<4, 16>(lds + OFF_WR, WR_STR, bb, acc, b, h);
#pragma unroll
  for (int t = 0; t < 4; ++t)
#pragma unroll
    for (int v = 0; v < 8; ++v) acc[t][v] = fmaxf(acc[t][v], 0.f) - sd[t][v];

  // f = softmax(Wout @ s^T) over the 32 obs of each row
  d2b<4>(sd, bb, h);
  v8f fa[2] = { {}, {} };
  wmma_mm<2, 16>(lds + OFF_WO, WO_STR, bb, fa, b, h);

  float mx = -3.4e38f;
#pragma unroll
  for (int t2 = 0; t2 < 2; ++t2)
#pragma unroll
    for (int v = 0; v < 8; ++v) mx = fmaxf(mx, fa[t2][v]);
  mx = fmaxf(mx, __shfl_xor(mx, 16, 32));
  float ssum = 0.f;
#pragma unroll
  for (int t2 = 0; t2 < 2; ++t2)
#pragma unroll
    for (int v = 0; v < 8; ++v) { float e = __expf(fa[t2][v] - mx); fa[t2][v] = e; ssum += e; }
  ssum += __shfl_xor(ssum, 16, 32);
  float inv = 1.f / ssum;
  float dot = 0.f;
#pragma unroll
  for (int t2 = 0; t2 < 2; ++t2)
#pragma unroll
    for (int v = 0; v < 8; ++v) {
      float f = fa[t2][v] * inv; fa[t2][v] = f;
      int oidx = 16*t2 + 8*h + v;
      dot += f * (((oidx == ob) ? 1.f : 0.f) - f);   // sum(f * eps_p)
    }
  dot += __shfl_xor(dot, 16, 32);
  float jp[2][8];
#pragma unroll
  for (int t2 = 0; t2 < 2; ++t2)
#pragma unroll
    for (int v = 0; v < 8; ++v) {
      float f = fa[t2][v];
      int oidx = 16*t2 + 8*h + v;
      jp[t2][v] = f * ((((oidx == ob) ? 1.f : 0.f) - f) - dot);
    }

  // acc += Wout^T @ jp^T   (K = 32)
  v2f bj[8];
  d2b<2>(jp, bj, h);
  wmma_mm<4, 8>(lds + OFF_WOT, WOT_STR, bj, acc, b, h);

  // new_s = norm(s + lr * acc); new_prev = s
  float ns[4][8];
  float ssq = 0.f;
#pragma unroll
  for (int t = 0; t < 4; ++t)
#pragma unroll
    for (int v = 0; v < 8; ++v) { float x = sd[t][v] + LR_ITER * acc[t][v]; ns[t][v] = x; ssq += x*x; }
  ssq += __shfl_xor(ssq, 16, 32);
  float sc = 1.f / (sqrtf(ssq) + EPS_N);
  float dsum = 0.f;
#pragma unroll
  for (int t = 0; t < 4; ++t)
#pragma unroll
    for (int v = 0; v < 8; ++v) { float x = ns[t][v] * sc; ns[t][v] = x; dsum += sd[t][v] - x; }

  store_state(pbuf, r, h, sd);
  store_state(sbuf, r, h, ns);
  block_accum(dsum, lds, partials);
}

// ---------------------------------------------------------------------------
__global__ void __launch_bounds__(NTHREADS) tpcn_final(
    const float* __restrict__ sbuf, const float* __restrict__ Wout, float* __restrict__ out)
{
  __shared__ float lds[LDS_TOT];
  for (int i = threadIdx.x; i < 32*64; i += NTHREADS) lds[OFF_WO + (i>>6)*WO_STR + (i&63)] = Wout[i];
  __syncthreads();

  int wave = threadIdx.x >> 5, lane = threadIdx.x & 31;
  int b = lane & 15, h = lane >> 4;
  int r = blockIdx.x * ROWS_PER_BLOCK + wave * 16 + b;

  float sd[4][8];
  load_state(sbuf, r, h, sd);
  v2f bb[16];
  d2b<4>(sd, bb, h);
  v8f fa[2] = { {}, {} };
  wmma_mm<2, 16>(lds + OFF_WO, WO_STR, bb, fa, b, h);

  float mx = -3.4e38f;
#pragma unroll
  for (int t2 = 0; t2 < 2; ++t2)
#pragma unroll
    for (int v = 0; v < 8; ++v) mx = fmaxf(mx, fa[t2][v]);
  mx = fmaxf(mx, __shfl_xor(mx, 16, 32));
  float ssum = 0.f;
#pragma unroll
  for (int t2 = 0; t2 < 2; ++t2)
#pragma unroll
    for (int v = 0; v < 8; ++v) { float e = __expf(fa[t2][v] - mx); fa[t2][v] = e; ssum += e; }
  ssum += __shfl_xor(ssum, 16, 32);
  float inv = 1.f / ssum;
#pragma unroll
  for (int t2 = 0; t2 < 2; ++t2)
#pragma unroll
    for (int v = 0; v < 8; ++v) fa[t2][v] *= inv;

#pragma unroll
  for (int t2 = 0; t2 < 2; ++t2) {
    float4* p4 = (float4*)(out + (size_t)r * NOBS + 16*t2 + 8*h);
    p4[0] = make_float4(fa[t2][0], fa[t2][1], fa[t2][2], fa[t2][3]);
    p4[1] = make_float4(fa[t2][4], fa[t2][5], fa[t2][6], fa[t2][7]);
  }
  float* outS = out + (size_t)B_TOTAL * NOBS;
  store_state(outS, r, h, sd);
}

// ---------------------------------------------------------------------------
extern "C" void kernel_launch(void* const* d_in, const int* in_sizes, int n_in,
                              void* d_out, int out_size, void* d_ws, size_t ws_size,
                              hipStream_t stream)
{
  const int*   dirs        = (const int*)  d_in[0];
  const int*   obs         = (const int*)  d_in[1];
  const float* prev_states = (const float*)d_in[2];
  const float* Wr          = (const float*)d_in[3];
  const float* Win         = (const float*)d_in[4];
  const float* Wout        = (const float*)d_in[5];

  float* sbuf     = (float*)d_ws;                       // B*64
  float* pbuf     = sbuf + (size_t)B_TOTAL * H_DIM;     // B*64
  float* partials = pbuf + (size_t)B_TOTAL * H_DIM;     // NBLOCKS
  float* meta     = partials + NBLOCKS;                 // [0]=done, [1..10]=active_i

  tpcn_init<<<NBLOCKS, NTHREADS, 0, stream>>>(dirs, prev_states, Wr, Win, sbuf, pbuf, partials);
  for (int it = 0; it < MAX_ITER; ++it) {
    tpcn_flag<<<1, NTHREADS, 0, stream>>>(meta, partials, it, it == 0 ? 1 : 0);
    tpcn_step<<<NBLOCKS, NTHREADS, 0, stream>>>(meta, it, sbuf, pbuf, dirs, obs,
                                                Wr, Wout, Win, partials);
  }
  tpcn_final<<<NBLOCKS, NTHREADS, 0, stream>>>(sbuf, Wout, (float*)d_out);

  (void)in_sizes; (void)n_in; (void)out_size; (void)ws_size;
}